// HGR_Network_56899726737499
// MI455X (gfx1250) — compile-verified
//
#include <hip/hip_runtime.h>
#include <hip/hip_bf16.h>

typedef __attribute__((ext_vector_type(2))) float v2f;
typedef __attribute__((ext_vector_type(8))) float v8f;
typedef __attribute__((ext_vector_type(4))) unsigned int u32x4;
typedef __attribute__((ext_vector_type(8))) int i32x8;
typedef __attribute__((ext_vector_type(4))) int i32x4;

#define NODES 4096
#define NBLK  1024
#define MMOD  4
#define INFEAT 128
#define HF    256
#define HF2   512

#define BT 64   // block tile (mask GEMM)
#define KT 32   // K panel staged via TDM

// ---------------------------------------------------------------------------
// V_WMMA_F32_16X16X4_F32 wrapper (neg_a, A, neg_b, B, c_mod, C, reuse_a, reuse_b)
// ---------------------------------------------------------------------------
__device__ __forceinline__ v8f wmma_f32_k4(v2f a, v2f b, v8f c) {
  return __builtin_amdgcn_wmma_f32_16x16x4_f32(false, a, false, b,
                                               (short)0, c, false, false);
}

// LDS byte offset of a shared-memory pointer (generic -> AS(3) -> int)
__device__ __forceinline__ unsigned lds_offset_of(const void* p) {
  return (unsigned)(unsigned long long)
         (const __attribute__((address_space(3))) void*)p;
}

// ---------------------------------------------------------------------------
// TDM: async 2-D f32 tile load Global -> LDS via Tensor Data Mover.
// D# packing per CDNA5 ISA 8.3/8.4: group0 {count=1, lds_addr, global_addr,
// type=2}; group1 {data_size=4B, tensor dims, tile dims, dim0 stride}.
// tile0 = contiguous extent (elements), tile1 = #lines, stride0 in elements.
// This toolchain exposes the 6-arg builtin (extra descriptor group + cpol).
// ---------------------------------------------------------------------------
__device__ __forceinline__ void tdm_load_2d_f32(const float* gptr, unsigned ldsOff,
                                                unsigned tile0, unsigned tile1,
                                                unsigned long long stride0,
                                                unsigned tdim0, unsigned tdim1) {
  unsigned long long ga = (unsigned long long)gptr;
  u32x4 g0;
  g0.x = 1u;                                              // count=1, user D#
  g0.y = ldsOff;                                          // lds_addr (bytes)
  g0.z = (unsigned)(ga & 0xFFFFFFFFull);                  // global_addr[31:0]
  g0.w = (unsigned)((ga >> 32) & 0x1FFFFFFull) | (2u << 30); // [56:32] | type=2
  i32x8 g1;
  g1[0] = (int)(2u << 16);                                // data_size = 4 bytes
  g1[1] = (int)((tdim0 & 0xFFFFu) << 16);                 // tensor_dim0[15:0]
  g1[2] = (int)((tdim0 >> 16) | ((tdim1 & 0xFFFFu) << 16));
  g1[3] = (int)((tdim1 >> 16) | (tile0 << 16));           // tile_dim0
  g1[4] = (int)tile1;                                     // tile_dim1 (dim2=0)
  g1[5] = (int)(stride0 & 0xFFFFFFFFull);                 // dim0_stride[31:0]
  g1[6] = (int)((stride0 >> 32) & 0xFFFFull);             // dim0_stride[47:32]
  g1[7] = 0;
  i32x4 z4 = {0, 0, 0, 0};
  i32x8 z8 = {0, 0, 0, 0, 0, 0, 0, 0};
  __builtin_amdgcn_tensor_load_to_lds(g0, g1, z4, z4, z8, 0);
}

// ---------------------------------------------------------------------------
// Center feature rows; rs[i] = 1/sqrt(sum_j xc^2) so R_ij = G_ij*rs_i*rs_j.
// ---------------------------------------------------------------------------
__global__ void k_prep(const float* __restrict__ f, float* __restrict__ xc,
                       float* __restrict__ rs) {
  int row = blockIdx.x;
  int t = threadIdx.x;            // 128 threads
  __shared__ float red[INFEAT];
  float v = f[(size_t)row * INFEAT + t];
  red[t] = v; __syncthreads();
  for (int s = 64; s > 0; s >>= 1) { if (t < s) red[t] += red[t + s]; __syncthreads(); }
  float mean = red[0] * (1.0f / INFEAT);
  __syncthreads();
  float x = v - mean;
  xc[(size_t)row * INFEAT + t] = x;
  red[t] = x * x; __syncthreads();
  for (int s = 64; s > 0; s >>= 1) { if (t < s) red[t] += red[t + s]; __syncthreads(); }
  if (t == 0) rs[row] = rsqrtf(red[0]);
}

// dr = sigmoid(sparse[12]); wsoft = softmax(c[4])
__global__ void k_sigsoft(const float* __restrict__ sparse, const float* __restrict__ c,
                          float* __restrict__ dr, float* __restrict__ wsoft) {
  int t = threadIdx.x;
  if (t < 12) dr[t] = 1.0f / (1.0f + expf(-sparse[t]));
  if (t == 0) {
    float m = c[0];
    for (int i = 1; i < 4; ++i) m = fmaxf(m, c[i]);
    float e[4], s = 0.f;
    for (int i = 0; i < 4; ++i) { e[i] = expf(c[i] - m); s += e[i]; }
    for (int i = 0; i < 4; ++i) wsoft[i] = e[i] / s;
  }
}

// ---------------------------------------------------------------------------
// 1024x1024 thresholded correlation block (one wave per 16x16 tile, K=128).
// dst[p,q] = relu(|clamp(G*rs_u*rs_v,-1,1)| - (u==v) - dr[drIdx])
// ---------------------------------------------------------------------------
__global__ void k_corr_block(const float* __restrict__ xc, const float* __restrict__ rs,
                             const float* __restrict__ dr, int drIdx,
                             int rowBlk, int colBlk, int subEye,
                             float* __restrict__ dst) {
  const int lane = threadIdx.x;
  const int half = lane >> 4, l15 = lane & 15;
  const int tm = blockIdx.y * 16, tn = blockIdx.x * 16;
  const float* aRow = xc + (size_t)(rowBlk * NBLK + tm + l15) * INFEAT;
  const float* bRow = xc + (size_t)(colBlk * NBLK + tn + l15) * INFEAT;
  v8f acc = {};
  for (int kk = 0; kk < INFEAT; kk += 4) {
    int k0 = kk + half * 2;
    v2f a = *(const v2f*)(aRow + k0);
    v2f b = *(const v2f*)(bRow + k0);
    acc = wmma_f32_k4(a, b, acc);
  }
  float drv = dr[drIdx];
#pragma unroll
  for (int r = 0; r < 8; ++r) {
    int m = r + half * 8, n = l15;
    int gu = rowBlk * NBLK + tm + m, gv = colBlk * NBLK + tn + n;
    float val = acc[r] * rs[gu] * rs[gv];
    val = fabsf(fminf(fmaxf(val, -1.f), 1.f));
    if (subEye && gu == gv) val -= 1.f;
    dst[(size_t)(tm + m) * NBLK + tn + n] = fmaxf(val - drv, 0.f);
  }
}

// ---------------------------------------------------------------------------
// mask block (i,j) = (A_i @ C_ij != 0).  128 threads = 4 waves, each wave a
// 32x32 register tile (4 accumulators); B K-panel staged into LDS by the TDM.
// ---------------------------------------------------------------------------
__global__ void k_mask_gemm(const float* __restrict__ A, const float* __restrict__ B,
                            int transB, float* __restrict__ mask, int rowOff, int colOff) {
  __shared__ float ldsB[KT * BT];
  const int lane = threadIdx.x & 31;
  const int wave = threadIdx.x >> 5;          // 0..3
  const int waveM = wave >> 1, waveN = wave & 1;
  const int half = lane >> 4, l15 = lane & 15;
  const int tm = blockIdx.y * BT + waveM * 32;   // wave 32x32 origin (local)
  const int tn = blockIdx.x * BT + waveN * 32;
  const int tnB = blockIdx.x * BT;               // block column base
  const unsigned ldsOff = lds_offset_of(&ldsB[0]);

  const float* aRow0 = A + (size_t)(tm + l15) * NBLK;
  const float* aRow1 = A + (size_t)(tm + 16 + l15) * NBLK;
  const int nl0 = waveN * 32 + l15, nl1 = nl0 + 16;   // cols within block tile

  v8f acc00 = {}, acc01 = {}, acc10 = {}, acc11 = {};
  for (int kp = 0; kp < NBLK; kp += KT) {
    __syncthreads();                       // previous panel fully consumed
    if (wave == 0) {
      if (transB) {                        // panel = S[tnB..+63][kp..+31]
        tdm_load_2d_f32(B + (size_t)tnB * NBLK + kp, ldsOff, KT, BT,
                        NBLK, NBLK, NBLK);
      } else {                             // panel = B[kp..+31][tnB..+63]
        tdm_load_2d_f32(B + (size_t)kp * NBLK + tnB, ldsOff, BT, KT,
                        NBLK, NBLK, NBLK);
      }
      __builtin_amdgcn_s_wait_tensorcnt(0);
    }
    __syncthreads();                       // panel visible to all waves
    for (int kk = 0; kk < KT; kk += 4) {
      int k0 = kk + half * 2;
      v2f a0 = *(const v2f*)(aRow0 + kp + k0);
      v2f a1 = *(const v2f*)(aRow1 + kp + k0);
      v2f b0, b1;
      if (transB) {                        // ldsB[n*KT + k]
        b0 = *(const v2f*)(&ldsB[nl0 * KT + k0]);
        b1 = *(const v2f*)(&ldsB[nl1 * KT + k0]);
      } else {                             // ldsB[k*BT + n]
        b0.x = ldsB[k0 * BT + nl0]; b0.y = ldsB[(k0 + 1) * BT + nl0];
        b1.x = ldsB[k0 * BT + nl1]; b1.y = ldsB[(k0 + 1) * BT + nl1];
      }
      acc00 = wmma_f32_k4(a0, b0, acc00);
      acc01 = wmma_f32_k4(a0, b1, acc01);
      acc10 = wmma_f32_k4(a1, b0, acc10);
      acc11 = wmma_f32_k4(a1, b1, acc11);
    }
  }
#pragma unroll
  for (int r = 0; r < 8; ++r) {
    int m = r + half * 8, n = l15;
    size_t row0 = (size_t)(rowOff + tm + m) * NODES + colOff;
    size_t row1 = (size_t)(rowOff + tm + 16 + m) * NODES + colOff;
    int c0 = tn + n, c1 = tn + 16 + n;
    mask[row0 + c0] = (acc00[r] != 0.f) ? 1.f : 0.f;
    mask[row0 + c1] = (acc01[r] != 0.f) ? 1.f : 0.f;
    mask[row1 + c0] = (acc10[r] != 0.f) ? 1.f : 0.f;
    mask[row1 + c1] = (acc11[r] != 0.f) ? 1.f : 0.f;
  }
}

// diag mask block: AC[i,i] = A_i @ I = A_i
__global__ void k_mask_diag(const float* __restrict__ Ablk, float* __restrict__ mask, int blk) {
  int idx = blockIdx.x * blockDim.x + threadIdx.x;   // 1024*1024
  int r = idx >> 10, c = idx & (NBLK - 1);
  mask[(size_t)(blk * NBLK + r) * NODES + blk * NBLK + c] = (Ablk[idx] != 0.f) ? 1.f : 0.f;
}

__global__ void k_colsum(const float* __restrict__ mask, float* __restrict__ indeg1) {
  int v = blockIdx.x * blockDim.x + threadIdx.x;
  float s = 0.f;
  for (int u = 0; u < NODES; ++u) s += mask[(size_t)u * NODES + v];
  indeg1[v] = s;
}

__global__ void k_rowsum(const float* __restrict__ mask, float* __restrict__ indeg2) {
  int wave = (blockIdx.x * blockDim.x + threadIdx.x) >> 5;   // one wave per row
  int lane = threadIdx.x & 31;
  const float* row = mask + (size_t)wave * NODES;
  float s = 0.f;
  for (int v = lane; v < NODES; v += 32) s += row[v];
  for (int off = 16; off > 0; off >>= 1) s += __shfl_down(s, off, 32);
  if (lane == 0) indeg2[wave] = s;
}

__global__ void k_norms(const float* __restrict__ indeg1, const float* __restrict__ indeg2,
                        float* __restrict__ norm1, float* __restrict__ c1,
                        float* __restrict__ norm2, float* __restrict__ c2) {
  int t = blockIdx.x * blockDim.x + threadIdx.x;
  float d1 = indeg1[t], d2 = indeg2[t];
  float n1 = d1 > 0.f ? rsqrtf(d1) : 0.f;
  float n2 = d2 > 0.f ? rsqrtf(d2) : 0.f;
  norm1[t] = n1; c1[t] = d1 > 0.f ? n1 / d1 : 0.f;
  norm2[t] = n2; c2[t] = d2 > 0.f ? n2 / d2 : 0.f;
}

// ---------------------------------------------------------------------------
// xout[m,f] = feat[m,f] + cscale[m] * sum_k maskOp[m,k]*kscale[k]*feat[k,f]
// transA=1: maskOp[m,k]=mask[k,m] (g1);  transA=0: mask[m,k] (g2).
// One wave per 32x32 output tile (4 accumulators).
// ---------------------------------------------------------------------------
__global__ void k_agg(const float* __restrict__ mask, const float* __restrict__ feat, int F,
                      const float* __restrict__ kscale, const float* __restrict__ cscale,
                      int transA, float* __restrict__ xout) {
  const int lane = threadIdx.x, half = lane >> 4, l15 = lane & 15;
  const int tm = blockIdx.y * 32, tn = blockIdx.x * 32;
  v8f acc00 = {}, acc01 = {}, acc10 = {}, acc11 = {};
  for (int kk = 0; kk < NODES; kk += 4) {
    int k0 = kk + half * 2;
    float s0 = kscale[k0], s1 = kscale[k0 + 1];
    v2f a0, a1;
    if (transA) {
      const float* c0 = mask + (size_t)k0 * NODES;
      const float* c1 = mask + (size_t)(k0 + 1) * NODES;
      a0.x = c0[tm + l15];      a0.y = c1[tm + l15];
      a1.x = c0[tm + 16 + l15]; a1.y = c1[tm + 16 + l15];
    } else {
      a0 = *(const v2f*)(mask + (size_t)(tm + l15) * NODES + k0);
      a1 = *(const v2f*)(mask + (size_t)(tm + 16 + l15) * NODES + k0);
    }
    a0.x *= s0; a0.y *= s1; a1.x *= s0; a1.y *= s1;
    const float* f0 = feat + (size_t)k0 * F;
    const float* f1 = feat + (size_t)(k0 + 1) * F;
    v2f b0, b1;
    b0.x = f0[tn + l15];      b0.y = f1[tn + l15];
    b1.x = f0[tn + 16 + l15]; b1.y = f1[tn + 16 + l15];
    acc00 = wmma_f32_k4(a0, b0, acc00);
    acc01 = wmma_f32_k4(a0, b1, acc01);
    acc10 = wmma_f32_k4(a1, b0, acc10);
    acc11 = wmma_f32_k4(a1, b1, acc11);
  }
#pragma unroll
  for (int r = 0; r < 8; ++r) {
    int m0 = tm + r + half * 8, m1 = m0 + 16;
    int n0 = tn + l15, n1 = n0 + 16;
    xout[(size_t)m0 * F + n0] = feat[(size_t)m0 * F + n0] + cscale[m0] * acc00[r];
    xout[(size_t)m0 * F + n1] = feat[(size_t)m0 * F + n1] + cscale[m0] * acc01[r];
    xout[(size_t)m1 * F + n0] = feat[(size_t)m1 * F + n0] + cscale[m1] * acc10[r];
    xout[(size_t)m1 * F + n1] = feat[(size_t)m1 * F + n1] + cscale[m1] * acc11[r];
  }
}

// out[m, colOff+n] = relu( x[m,:] @ W[:,n] + bias[n] ),  W is [Fin][256]
__global__ void k_dense(const float* __restrict__ x, int Fin,
                        const float* __restrict__ W, const float* __restrict__ bias,
                        float* __restrict__ out, int ldOut, int colOff) {
  const int lane = threadIdx.x, half = lane >> 4, l15 = lane & 15;
  const int tm = blockIdx.y * 16, tn = blockIdx.x * 16;
  const float* xRow = x + (size_t)(tm + l15) * Fin;
  v8f acc = {};
  for (int kk = 0; kk < Fin; kk += 4) {
    int k0 = kk + half * 2;
    v2f a = *(const v2f*)(xRow + k0);
    v2f b;
    b.x = W[(size_t)k0 * HF + tn + l15];
    b.y = W[(size_t)(k0 + 1) * HF + tn + l15];
    acc = wmma_f32_k4(a, b, acc);
  }
#pragma unroll
  for (int r = 0; r < 8; ++r) {
    int m = tm + r + half * 8, n = tn + l15;
    out[(size_t)m * ldOut + colOff + n] = fmaxf(acc[r] + bias[n], 0.f);
  }
}

__global__ void k_bnstats(const float* __restrict__ h, float* __restrict__ mu,
                          float* __restrict__ var) {
  int col = blockIdx.x * blockDim.x + threadIdx.x;   // 512 cols
  float s = 0.f, sq = 0.f;
  for (int r = 0; r < NODES; ++r) {
    float v = h[(size_t)r * HF2 + col]; s += v; sq += v * v;
  }
  float m = s * (1.0f / NODES);
  mu[col] = m;
  var[col] = sq * (1.0f / NODES) - m * m;   // biased var (training-mode BN)
}

__global__ void k_bnapply(float* __restrict__ h, const float* __restrict__ mu,
                          const float* __restrict__ var, const float* __restrict__ gamma,
                          const float* __restrict__ beta) {
  int idx = blockIdx.x * blockDim.x + threadIdx.x;
  int col = idx & (HF2 - 1);
  h[idx] = (h[idx] - mu[col]) * rsqrtf(var[col] + 1e-5f) * gamma[col] + beta[col];
}

// out[node, cls] = sum_m wsoft[m] * (h[m*1024+node,:] @ w_ro[:,cls])
__global__ void k_readout(const float* __restrict__ h, const float* __restrict__ wro,
                          const float* __restrict__ wsoft, float* __restrict__ out) {
  int t = blockIdx.x * blockDim.x + threadIdx.x;   // 0..4095
  int node = t >> 2, cls = t & 3;
  float acc = 0.f;
  for (int m = 0; m < MMOD; ++m) {
    const float* row = h + (size_t)(m * NBLK + node) * HF2;
    float d = 0.f;
    for (int k = 0; k < HF2; ++k) d += row[k] * wro[k * 4 + cls];
    acc += wsoft[m] * d;
  }
  out[t] = acc;
}

// ---------------------------------------------------------------------------
extern "C" void kernel_launch(void* const* d_in, const int* in_sizes, int n_in,
                              void* d_out, int out_size, void* d_ws, size_t ws_size,
                              hipStream_t stream) {
  (void)in_sizes; (void)n_in; (void)out_size; (void)ws_size;
  const float* features = (const float*)d_in[0];
  const float* sparse   = (const float*)d_in[1];
  const float* cmix     = (const float*)d_in[2];
  const float* wAC1 = (const float*)d_in[3];  const float* bAC1 = (const float*)d_in[4];
  const float* wCA1 = (const float*)d_in[5];  const float* bCA1 = (const float*)d_in[6];
  const float* wAC2 = (const float*)d_in[7];  const float* bAC2 = (const float*)d_in[8];
  const float* wCA2 = (const float*)d_in[9];  const float* bCA2 = (const float*)d_in[10];
  const float* gamma1 = (const float*)d_in[11]; const float* beta1 = (const float*)d_in[12];
  const float* gamma2 = (const float*)d_in[13]; const float* beta2 = (const float*)d_in[14];
  const float* w_ro = (const float*)d_in[15];
  float* out = (float*)d_out;

  float* ws = (float*)d_ws;
  size_t off = 0;
  auto alloc = [&](size_t n) { float* p = ws + off; off += n; return p; };
  float* xc    = alloc((size_t)NODES * INFEAT);
  float* rs    = alloc(NODES);
  float* dr    = alloc(12);
  float* wsoft = alloc(4);
  float* Ab[4]; for (int i = 0; i < 4; ++i) Ab[i] = alloc((size_t)NBLK * NBLK);
  float* S4 = alloc((size_t)NBLK * NBLK);
  float* S7 = alloc((size_t)NBLK * NBLK);
  float* S9 = alloc((size_t)NBLK * NBLK);
  float* mask   = alloc((size_t)NODES * NODES);
  float* indeg1 = alloc(NODES); float* indeg2 = alloc(NODES);
  float* norm1  = alloc(NODES); float* c1v    = alloc(NODES);
  float* norm2  = alloc(NODES); float* c2v    = alloc(NODES);
  float* mu  = alloc(HF2);      float* var    = alloc(HF2);
  float* xA  = alloc((size_t)NODES * HF2);
  float* xB  = alloc((size_t)NODES * HF2);
  float* h1  = alloc((size_t)NODES * HF2);
  float* h2  = alloc((size_t)NODES * HF2);

  dim3 b32(32), g64(64, 64);

  k_prep<<<NODES, INFEAT, 0, stream>>>(features, xc, rs);
  k_sigsoft<<<1, 32, 0, stream>>>(sparse, cmix, dr, wsoft);

  // Diag A blocks use dr[1],dr[5],dr[8],dr[10]; live C-source blocks (per the
  // reference's stale-blk trace) are (0,3)/dr4, (1,3)/dr7, (2,3)/dr9.
  const int drA[4] = {1, 5, 8, 10};
  for (int i = 0; i < 4; ++i)
    k_corr_block<<<g64, b32, 0, stream>>>(xc, rs, dr, drA[i], i, i, 1, Ab[i]);
  k_corr_block<<<g64, b32, 0, stream>>>(xc, rs, dr, 4, 0, 3, 0, S4);
  k_corr_block<<<g64, b32, 0, stream>>>(xc, rs, dr, 7, 1, 3, 0, S7);
  k_corr_block<<<g64, b32, 0, stream>>>(xc, rs, dr, 9, 2, 3, 0, S9);

  // mask = (A @ C != 0), blockwise (A is block-diagonal)
  for (int i = 0; i < 4; ++i)
    k_mask_diag<<<(NBLK * NBLK) / 256, 256, 0, stream>>>(Ab[i], mask, i);
  struct MaskJob { int i, j; const float* B; int tr; };
  MaskJob jobs[12] = {
    {0, 1, S4, 1}, {0, 2, S7, 1}, {0, 3, S9, 1},
    {1, 0, S4, 0}, {1, 2, S7, 1}, {1, 3, S9, 1},
    {2, 0, S7, 0}, {2, 1, S7, 0}, {2, 3, S9, 1},
    {3, 0, S9, 0}, {3, 1, S9, 0}, {3, 2, S9, 0},
  };
  dim3 gMask(NBLK / BT, NBLK / BT);     // 16x16 blocks of 128 threads
  for (int t = 0; t < 12; ++t)
    k_mask_gemm<<<gMask, 128, 0, stream>>>(Ab[jobs[t].i], jobs[t].B, jobs[t].tr,
                                           mask, jobs[t].i * NBLK, jobs[t].j * NBLK);

  k_colsum<<<NODES / 256, 256, 0, stream>>>(mask, indeg1);
  k_rowsum<<<NODES / 8, 256, 0, stream>>>(mask, indeg2);   // 1 wave per row
  k_norms<<<NODES / 256, 256, 0, stream>>>(indeg1, indeg2, norm1, c1v, norm2, c2v);

  // Layer 1 (F=128)
  k_agg<<<dim3(INFEAT / 32, NODES / 32), b32, 0, stream>>>(mask, features, INFEAT, norm1, c1v, 1, xA);
  k_agg<<<dim3(INFEAT / 32, NODES / 32), b32, 0, stream>>>(mask, features, INFEAT, norm2, c2v, 0, xB);
  k_dense<<<dim3(HF / 16, NODES / 16), b32, 0, stream>>>(xA, INFEAT, wAC1, bAC1, h1, HF2, 0);
  k_dense<<<dim3(HF / 16, NODES / 16), b32, 0, stream>>>(xB, INFEAT, wCA1, bCA1, h1, HF2, HF);
  k_bnstats<<<HF2 / 256, 256, 0, stream>>>(h1, mu, var);
  k_bnapply<<<(NODES * HF2) / 256, 256, 0, stream>>>(h1, mu, var, gamma1, beta1);

  // Layer 2 (F=512)
  k_agg<<<dim3(HF2 / 32, NODES / 32), b32, 0, stream>>>(mask, h1, HF2, norm1, c1v, 1, xA);
  k_agg<<<dim3(HF2 / 32, NODES / 32), b32, 0, stream>>>(mask, h1, HF2, norm2, c2v, 0, xB);
  k_dense<<<dim3(HF / 16, NODES / 16), b32, 0, stream>>>(xA, HF2, wAC2, bAC2, h2, HF2, 0);
  k_dense<<<dim3(HF / 16, NODES / 16), b32, 0, stream>>>(xB, HF2, wCA2, bCA2, h2, HF2, HF);
  k_bnstats<<<HF2 / 256, 256, 0, stream>>>(h2, mu, var);
  k_bnapply<<<(NODES * HF2) / 256, 256, 0, stream>>>(h2, mu, var, gamma2, beta2);

  k_readout<<<NODES / 256, 256, 0, stream>>>(h2, w_ro, wsoft, out);
}